// GINConv_49795850829912
// MI455X (gfx1250) — compile-verified
//
#include <hip/hip_runtime.h>
#include <hip/hip_bf16.h>

#define N_NODES 50000
#define N_EDGES 800000
#define DFEAT   64
#define NBLK    (N_NODES / 16)   // 3125 row blocks of 16 nodes

typedef __attribute__((ext_vector_type(2))) float v2f;
typedef __attribute__((ext_vector_type(8))) float v8f;

// ---------------------------------------------------------------------------
// Kernel 1: y = (1 + eps) * x  (GIN self term), and zero the BN stat buffer.
// ---------------------------------------------------------------------------
__global__ void gin_init_kernel(const float* __restrict__ x,
                                const float* __restrict__ eps,
                                float* __restrict__ y,
                                float* __restrict__ stats)
{
    int i = blockIdx.x * blockDim.x + threadIdx.x;
    if (i < N_NODES * DFEAT) {
        y[i] = (1.0f + eps[0]) * x[i];
    }
    if (i < 128) {
        stats[i] = 0.0f;   // [0..63] = sum, [64..127] = sumsq
    }
}

// ---------------------------------------------------------------------------
// Kernel 2: edge scatter-add:  y[dst] += x[src] * w.
// One wave per edge; lane l handles features {2l, 2l+1} -> contiguous float2
// gather (256B/wave) and contiguous f32 atomics. x and y are L2-resident.
// ---------------------------------------------------------------------------
__global__ void gin_aggregate_kernel(const float* __restrict__ x,
                                     const int*   __restrict__ src,
                                     const int*   __restrict__ dst,
                                     const float* __restrict__ ew,
                                     float* __restrict__ y)
{
    int tid  = blockIdx.x * blockDim.x + threadIdx.x;
    int e    = tid >> 5;
    int lane = tid & 31;
    if (e >= N_EDGES) return;

    int   s = src[e];
    int   d = dst[e];
    float w = ew[e];

    const v2f v = *(const v2f*)(x + s * DFEAT + 2 * lane);
    float* yp = y + d * DFEAT + 2 * lane;
    atomicAdd(yp,     v.x * w);
    atomicAdd(yp + 1, v.y * w);
}

// ---------------------------------------------------------------------------
// Kernels 3/4: Hout = act(Hin @ W^T + bias), W is [64,64] row-major [out,in].
// One wave computes a 16-node x 64-feature tile with v_wmma_f32_16x16x4_f32.
//   A (16x4 f32):  lane L -> M = L%16, K = 2*(L/16) + {0,1}   (float2 load)
//   B (4x16 f32):  lane L -> N = L%16, K = 2*(L/16) + {0,1}   (float2 load)
//   C/D (16x16):   vgpr r, lane L -> M = r + 8*(L/16), N = L%16
// DO_STATS: per-feature sum/sumsq reduced across the wave (xor-16 shuffle)
// and atomically accumulated for BatchNorm.
// ---------------------------------------------------------------------------
template <bool DO_RELU, bool DO_STATS>
__global__ void gin_mlp_kernel(const float* __restrict__ Hin,
                               const float* __restrict__ W,
                               const float* __restrict__ bias,
                               float* __restrict__ Hout,
                               float* __restrict__ stats)
{
    const int wave = blockIdx.x * (blockDim.x >> 5) + (threadIdx.x >> 5);
    if (wave >= NBLK) return;              // wave-uniform guard: EXEC stays all-1s
    const int lane  = threadIdx.x & 31;
    const int lhalf = lane >> 4;           // 0 or 1
    const int lmod  = lane & 15;
    const int mbase = wave * 16;

    const float* arow = Hin + (mbase + lmod) * DFEAT + 2 * lhalf;

    v8f acc[4];
#pragma unroll
    for (int nt = 0; nt < 4; ++nt) {
        const float b = bias[nt * 16 + lmod];          // feature j = nt*16 + lmod
        acc[nt] = (v8f){b, b, b, b, b, b, b, b};
    }

#pragma unroll
    for (int ks = 0; ks < 16; ++ks) {
        const int k0 = ks * 4;
        const v2f a = *(const v2f*)(arow + k0);        // k = k0 + 2*lhalf + {0,1}
#pragma unroll
        for (int nt = 0; nt < 4; ++nt) {
            const v2f bm = *(const v2f*)(W + (nt * 16 + lmod) * DFEAT + k0 + 2 * lhalf);
            acc[nt] = __builtin_amdgcn_wmma_f32_16x16x4_f32(
                /*neg_a=*/false, a, /*neg_b=*/false, bm,
                /*c_mod=*/(short)0, acc[nt],
                /*reuse_a=*/false, /*reuse_b=*/false);
        }
    }

#pragma unroll
    for (int nt = 0; nt < 4; ++nt) {
        float s = 0.0f, s2 = 0.0f;
#pragma unroll
        for (int r = 0; r < 8; ++r) {
            float v = acc[nt][r];
            if (DO_RELU) v = fmaxf(v, 0.0f);
            Hout[(mbase + r + 8 * lhalf) * DFEAT + nt * 16 + lmod] = v;
            if (DO_STATS) { s += v; s2 += v * v; }
        }
        if (DO_STATS) {
            s  += __shfl_xor(s, 16, 32);               // combine the two M-halves
            s2 += __shfl_xor(s2, 16, 32);
            if (lhalf == 0) {                          // one lane per feature
                atomicAdd(&stats[nt * 16 + lmod],      s);
                atomicAdd(&stats[64 + nt * 16 + lmod], s2);
            }
        }
    }
}

// ---------------------------------------------------------------------------
// Kernel 5: BatchNorm (training stats, biased var) + ReLU.
// ---------------------------------------------------------------------------
__global__ void gin_bn_relu_kernel(const float* __restrict__ h,
                                   const float* __restrict__ stats,
                                   const float* __restrict__ gamma,
                                   const float* __restrict__ beta,
                                   float* __restrict__ out)
{
    int i = blockIdx.x * blockDim.x + threadIdx.x;
    if (i >= N_NODES * DFEAT) return;
    const int j = i & (DFEAT - 1);
    const float invN = 1.0f / (float)N_NODES;
    const float mean = stats[j] * invN;
    const float var  = stats[64 + j] * invN - mean * mean;
    const float sc   = rsqrtf(var + 1e-5f) * gamma[j];
    const float v    = (h[i] - mean) * sc + beta[j];
    out[i] = fmaxf(v, 0.0f);
}

// ---------------------------------------------------------------------------
extern "C" void kernel_launch(void* const* d_in, const int* in_sizes, int n_in,
                              void* d_out, int out_size, void* d_ws, size_t ws_size,
                              hipStream_t stream)
{
    const float* x     = (const float*)d_in[0];
    const int*   src   = (const int*)  d_in[1];
    const int*   dst   = (const int*)  d_in[2];
    const float* ew    = (const float*)d_in[3];
    const float* W1    = (const float*)d_in[4];
    const float* b1    = (const float*)d_in[5];
    const float* W2    = (const float*)d_in[6];
    const float* b2    = (const float*)d_in[7];
    const float* eps   = (const float*)d_in[8];
    const float* gamma = (const float*)d_in[9];
    const float* beta  = (const float*)d_in[10];
    float* out = (float*)d_out;

    const size_t NF = (size_t)N_NODES * DFEAT;        // 3,200,000 floats
    float* y     = (float*)d_ws;                      // aggregation buffer
    float* h1    = y  + NF;                           // after layer 1
    float* h2    = h1 + NF;                           // after layer 2 (pre-BN)
    float* stats = h2 + NF;                           // 128 floats: sum / sumsq

    // 1) y = (1+eps)*x ; zero stats
    gin_init_kernel<<<(int)((NF + 255) / 256), 256, 0, stream>>>(x, eps, y, stats);

    // 2) y[dst] += x[src] * w   (one wave per edge)
    {
        const long long threads = (long long)N_EDGES * 32;
        gin_aggregate_kernel<<<(int)((threads + 255) / 256), 256, 0, stream>>>(
            x, src, dst, ew, y);
    }

    // 3) h1 = relu(y @ W1^T + b1)
    const int waves_per_blk = 256 / 32;
    const int gemm_blocks   = (NBLK + waves_per_blk - 1) / waves_per_blk;
    gin_mlp_kernel<true,  false><<<gemm_blocks, 256, 0, stream>>>(y,  W1, b1, h1, stats);

    // 4) h2 = h1 @ W2^T + b2, with fused per-feature sum/sumsq for BN
    gin_mlp_kernel<false, true ><<<gemm_blocks, 256, 0, stream>>>(h1, W2, b2, h2, stats);

    // 5) out = relu(BN(h2))
    gin_bn_relu_kernel<<<(int)((NF + 255) / 256), 256, 0, stream>>>(
        h2, stats, gamma, beta, out);
}